// KPConv_82970178224387
// MI455X (gfx1250) — compile-verified
//
#include <hip/hip_runtime.h>

// KPConv as a single f32 GEMM:  out[p, o] = sum_{c=0..47} A[p,c] * W[c,o]
//   A[p, k*3+i] = exp(-0.5*|x_p - kp_k|^2 / sigma^2) * x[p,i]
//   W[k*3+i, o] = weights[k,i,o]   (contiguous reinterpret)
//
// Memory-bound: 134 MB of output stores vs 3.2 GFLOP of math -> keep f32
// precision and use V_WMMA_F32_16X16X4_F32 (K=4 steps, 12 per 16x16 tile).
//
// Wave mapping (wave32): 16 points (M-tile) x 64 channels (4 N-tiles),
// 48 WMMAs per wave across 4 independent accumulator chains.

typedef __attribute__((ext_vector_type(2))) float v2f;
typedef __attribute__((ext_vector_type(8))) float v8f;

#define KPTS 16
#define CIN  3
#define COUT 64
#define KC   (KPTS * CIN)       // 48 = GEMM K dimension
#define WAVES_PER_BLOCK 8
#define PTS_PER_WAVE    16
#define BLOCK_THREADS   256

__launch_bounds__(BLOCK_THREADS, 2)
__global__ void kpconv_wmma_f32(const float* __restrict__ x,       // [P,3]
                                const float* __restrict__ kp,      // [16,3]
                                const float* __restrict__ w,       // [16,3,64] == [48,64]
                                const float* __restrict__ sigma_p, // [1]
                                float* __restrict__ out)           // [P,64]
{
    __shared__ float wlds[KC * COUT];   // 12 KB staged weights

    const int tid = threadIdx.x;

    // Cooperative stage of the 48x64 weight matrix into LDS (12 dwords/thread).
    #pragma unroll
    for (int i = 0; i < (KC * COUT) / BLOCK_THREADS; ++i)
        wlds[tid + i * BLOCK_THREADS] = w[tid + i * BLOCK_THREADS];

    const int lane = tid & 31;
    const int wv   = tid >> 5;
    const int row  = lane & 15;          // matrix row (point within wave tile)
    const bool hi  = lane >= 16;         // upper lane half

    const int wgid = blockIdx.x * WAVES_PER_BLOCK + wv;   // global wave id
    const int p    = wgid * PTS_PER_WAVE + row;           // this lane's point

    // Load point coords (both lane halves load the same point -> L0 hit).
    float xv0 = x[p * 3 + 0];
    float xv1 = x[p * 3 + 1];
    float xv2 = x[p * 3 + 2];

    const float sg   = sigma_p[0];
    const float coef = -0.5f / (sg * sg);

    // Gaussian correlation to each of the 16 kernel points (v_exp_f32).
    float g[KPTS];
    #pragma unroll
    for (int k = 0; k < KPTS; ++k) {
        float dx = xv0 - kp[k * 3 + 0];   // uniform -> scalar loads
        float dy = xv1 - kp[k * 3 + 1];
        float dz = xv2 - kp[k * 3 + 2];
        g[k] = __expf(coef * (dx * dx + dy * dy + dz * dz));
    }

    // Build A fragments for the 12 K-steps.
    // ISA 32-bit A 16x4 layout: VGPR0 = {K+0 (lanes 0-15), K+2 (lanes 16-31)},
    //                           VGPR1 = {K+1, K+3}.
    // All indices compile-time; lane-half resolved with selects (no v-indexing).
    v2f a[12];
    float xvv[3] = {xv0, xv1, xv2};
    #pragma unroll
    for (int s = 0; s < 12; ++s) {
        const int c = 4 * s;
        float p0 = g[(c + 0) / 3] * xvv[(c + 0) % 3];
        float p1 = g[(c + 1) / 3] * xvv[(c + 1) % 3];
        float p2 = g[(c + 2) / 3] * xvv[(c + 2) % 3];
        float p3 = g[(c + 3) / 3] * xvv[(c + 3) % 3];
        a[s].x = hi ? p2 : p0;
        a[s].y = hi ? p3 : p1;
    }

    __syncthreads();   // weights staged

    // Four independent 16x16 accumulators (channels 0-15,16-31,32-47,48-63).
    v8f c0 = {}, c1 = {}, c2 = {}, c3 = {};
    const int cb = hi ? 2 : 0;   // B 4x16 layout: halves split at rows/2

    #pragma unroll
    for (int s = 0; s < 12; ++s) {
        const int kr = 4 * s + cb;
        v2f b0, b1, b2, b3;
        b0.x = wlds[kr * COUT +  0 + row];  b0.y = wlds[(kr + 1) * COUT +  0 + row];
        b1.x = wlds[kr * COUT + 16 + row];  b1.y = wlds[(kr + 1) * COUT + 16 + row];
        b2.x = wlds[kr * COUT + 32 + row];  b2.y = wlds[(kr + 1) * COUT + 32 + row];
        b3.x = wlds[kr * COUT + 48 + row];  b3.y = wlds[(kr + 1) * COUT + 48 + row];

        c0 = __builtin_amdgcn_wmma_f32_16x16x4_f32(false, a[s], false, b0, (short)0, c0, false, false);
        c1 = __builtin_amdgcn_wmma_f32_16x16x4_f32(false, a[s], false, b1, (short)0, c1, false, false);
        c2 = __builtin_amdgcn_wmma_f32_16x16x4_f32(false, a[s], false, b2, (short)0, c2, false, false);
        c3 = __builtin_amdgcn_wmma_f32_16x16x4_f32(false, a[s], false, b3, (short)0, c3, false, false);
    }

    // Store: C/D layout -> VGPR r holds (M=r, N=lane) / (M=r+8, N=lane-16).
    // Each half-wave writes a contiguous 64B-aligned 64B chunk per store.
    const int outbase = wgid * (PTS_PER_WAVE * COUT);
    #pragma unroll
    for (int r = 0; r < 8; ++r) {
        const int m = r + (hi ? 8 : 0);
        const int o = outbase + m * COUT + row;
        out[o +  0] = c0[r];
        out[o + 16] = c1[r];
        out[o + 32] = c2[r];
        out[o + 48] = c3[r];
    }
}

extern "C" void kernel_launch(void* const* d_in, const int* in_sizes, int n_in,
                              void* d_out, int out_size, void* d_ws, size_t ws_size,
                              hipStream_t stream) {
    const float* x  = (const float*)d_in[0];   // [B,N,3] f32
    const float* kp = (const float*)d_in[1];   // [16,3]  f32
    const float* w  = (const float*)d_in[2];   // [16,3,64] f32
    const float* sg = (const float*)d_in[3];   // [1] f32
    float* out = (float*)d_out;                // [B,N,64] f32

    const int P = in_sizes[0] / CIN;                       // B*N = 524288
    const int blocks = P / (PTS_PER_WAVE * WAVES_PER_BLOCK); // 4096 (exact)

    kpconv_wmma_f32<<<blocks, BLOCK_THREADS, 0, stream>>>(x, kp, w, sg, out);
}